// AMSE_3367254360372
// MI455X (gfx1250) — compile-verified
//
#include <hip/hip_runtime.h>
#include <math.h>

#define Bdim 64
#define Fdim 32
#define Sdim 512
#define Pdim 96
#define Edim 8
#define Hdim 2048
#define FB   (Fdim * Bdim)        // 2048
#define OUTN (Bdim * Fdim * Pdim) // 196608

typedef float v2f __attribute__((ext_vector_type(2)));
typedef float v8f __attribute__((ext_vector_type(8)));

__device__ __forceinline__ v8f wmma_f32x4(v2f a, v2f b, v8f c) {
  // D = A(16x4,f32) * B(4x16,f32) + C(16x16,f32)
  return __builtin_amdgcn_wmma_f32_16x16x4_f32(false, a, false, b, (short)0, c,
                                               false, false);
}

// ---------------------------------------------------------------------------
// 1) Gating: logits = tf @ Wg + bg ; top-2 threshold ; double softmax
//    One wave32 per (f,b) row.  gates[row*8+e] written to workspace.
// ---------------------------------------------------------------------------
__global__ __launch_bounds__(256) void gates_kernel(
    const float* __restrict__ te, const float* __restrict__ Wg,
    const float* __restrict__ bg, float* __restrict__ gates) {
  const int wave = threadIdx.x >> 5;
  const int lane = threadIdx.x & 31;
  const int row  = blockIdx.x * 8 + wave;      // row = f*B + b
  const int f = row >> 6, b = row & 63;
  const float* trow = te + (size_t)(b * Fdim + f) * Sdim;

  float acc[8];
#pragma unroll
  for (int e = 0; e < 8; ++e) acc[e] = 0.f;
  for (int s = lane; s < Sdim; s += 32) {
    const float t = trow[s];
    const float4 w0 = ((const float4*)(Wg + s * 8))[0];
    const float4 w1 = ((const float4*)(Wg + s * 8))[1];
    acc[0] += t * w0.x; acc[1] += t * w0.y; acc[2] += t * w0.z; acc[3] += t * w0.w;
    acc[4] += t * w1.x; acc[5] += t * w1.y; acc[6] += t * w1.z; acc[7] += t * w1.w;
  }
#pragma unroll
  for (int e = 0; e < 8; ++e) {
#pragma unroll
    for (int off = 16; off >= 1; off >>= 1)
      acc[e] += __shfl_xor(acc[e], off, 32);
  }
  float lg[8];
  float m1 = -1e30f, m2 = -1e30f;
#pragma unroll
  for (int e = 0; e < 8; ++e) {
    lg[e] = acc[e] + bg[e];
    const float v = lg[e];
    if (v > m1) { m2 = m1; m1 = v; } else if (v > m2) { m2 = v; }
  }
  float se = 0.f, ex[8];
#pragma unroll
  for (int e = 0; e < 8; ++e) { ex[e] = expf(lg[e] - m1); se += ex[e]; }
  const float inv = 1.f / se;
  float dec[8];
  float dmax = -1e30f;
#pragma unroll
  for (int e = 0; e < 8; ++e) {
    const float sm = ex[e] * inv;
    dec[e] = (lg[e] < m2) ? 10.f * logf(sm + 1.f) : 10.f * (expf(sm) - 1.f);
    dmax = fmaxf(dmax, dec[e]);
  }
  float ds = 0.f;
#pragma unroll
  for (int e = 0; e < 8; ++e) { dec[e] = expf(dec[e] - dmax); ds += dec[e]; }
  const float dinv = 1.f / ds;
  if (lane == 0) {
#pragma unroll
    for (int e = 0; e < 8; ++e) gates[row * 8 + e] = dec[e] * dinv;
  }
}

// ---------------------------------------------------------------------------
// 2) out[b,f,p] = sum_e gates[f,b,e] * b2[e,p]   (seed for atomic accum)
// ---------------------------------------------------------------------------
__global__ __launch_bounds__(256) void init_kernel(
    const float* __restrict__ gates, const float* __restrict__ b2,
    float* __restrict__ out) {
  const int idx = blockIdx.x * 256 + threadIdx.x;     // < OUTN exactly
  const int p = idx % Pdim;
  const int bf = idx / Pdim;
  const int b = bf / Fdim, f = bf % Fdim;
  const int r = f * Bdim + b;
  float s = 0.f;
#pragma unroll
  for (int e = 0; e < 8; ++e) s += gates[r * 8 + e] * b2[e * Pdim + p];
  out[idx] = s;
}

// ---------------------------------------------------------------------------
// 3) selector / entropy losses -> d_out[OUTN], d_out[OUTN+1]
// ---------------------------------------------------------------------------
__global__ __launch_bounds__(256) void loss_kernel(
    const float* __restrict__ gates, float* __restrict__ out) {
  __shared__ float gs[Fdim][Edim];
  __shared__ float cvs[Fdim];
  __shared__ float ents[Fdim];
  const int t = threadIdx.x;
  const int f = t >> 3, e = t & 7;
  float s = 0.f;
  for (int b = 0; b < Bdim; ++b) s += gates[(f * Bdim + b) * 8 + e];
  gs[f][e] = s;
  __syncthreads();
  if (t < Fdim) {
    float m = 0.f;
#pragma unroll
    for (int k = 0; k < 8; ++k) m += gs[t][k];
    m *= (1.f / 8.f);
    float ss = 0.f, ent = 0.f;
#pragma unroll
    for (int k = 0; k < 8; ++k) {
      const float d = gs[t][k] - m;
      ss += d * d;
      const float gm = gs[t][k] * (1.f / (float)Bdim);
      ent += -gm * logf(gm + 1e-8f);
    }
    const float var = (float)Pdim * ss / (float)(Edim * Pdim - 1);
    cvs[t] = var / (m * m + 1e-10f);
    ents[t] = ent * (1.f / 8.f);
  }
  __syncthreads();
  if (t == 0) {
    float a = 0.f, c = 0.f;
    for (int i = 0; i < Fdim; ++i) { a += cvs[i]; c += ents[i]; }
    out[OUTN] = a;
    out[OUTN + 1] = c;
  }
}

// ---------------------------------------------------------------------------
// 4) Fused MoE:  h = gelu(X@W1[e]+b1) ; out += gates * (h@W2[e])
//    grid (FB/64, E, H/512); block 256 (8 wave32s).
//    fp32 WMMA 16x16x4 for both GEMMs.
//    X tile staged via GLOBAL_LOAD_ASYNC_TO_LDS_B128 (ASYNCcnt);
//    h re-layout C->A through padded LDS.  h-step = 64: each wave owns one
//    16-col h-group and 2 M row-tiles -> W1 fragment reused x2; waves w and w+4
//    load the same W1 columns (1 global dword per WMMA in GEMM1).
// ---------------------------------------------------------------------------
__global__ __launch_bounds__(256) void moe_kernel(
    const float* __restrict__ x,  const float* __restrict__ W1,
    const float* __restrict__ b1, const float* __restrict__ W2,
    const float* __restrict__ b2, const float* __restrict__ gates,
    float* __restrict__ out) {
  __shared__ float Xs[64][Sdim + 4];   // stride 516 (== 4 mod 64): conflict-free
  __shared__ float Hs[64][68];         // stride 68  (== 4 mod 64): conflict-free

  const int tid  = threadIdx.x;
  const int wave = tid >> 5;
  const int lane = tid & 31;
  const int lane15 = lane & 15;
  const int khalf  = (lane >> 4) * 2;  // 0 or 2: A/B K sub-pair per half-wave
  const int r0 = blockIdx.x * 64;
  const int e  = blockIdx.y;
  const int h0 = blockIdx.z * 512;

  // ---- async-stage X tile: rows r0..r0+63 of xf (xf[f,b,s] = x[b,f,s]) ----
  for (int idx = tid; idx < 64 * (Sdim / 4); idx += 256) {
    const int i = idx >> 7;            // 128 float4 per row
    const int c = idx & 127;
    const int r = r0 + i;
    const int fb = (r & 63) * Fdim + (r >> 6);
    const float* gsrc = x + (size_t)fb * Sdim + c * 4;
    const unsigned ldst = (unsigned)(uintptr_t)&Xs[i][c * 4];
    asm volatile("global_load_async_to_lds_b128 %0, %1, off"
                 :: "v"(ldst), "v"(gsrc) : "memory");
  }
  asm volatile("s_wait_asynccnt 0x0" ::: "memory");
  __syncthreads();

  const int hg  = wave & 3;            // GEMM1 h-group (16 cols of 64)
  const int rtb = (wave >> 2) * 2;     // GEMM1 row-tile pair base
  const int rt  = wave & 3;            // GEMM2 row tile
  const int ptb = wave >> 2;           // GEMM2 P-tile base

  v8f acc[3] = {{0.f,0.f,0.f,0.f,0.f,0.f,0.f,0.f},
                {0.f,0.f,0.f,0.f,0.f,0.f,0.f,0.f},
                {0.f,0.f,0.f,0.f,0.f,0.f,0.f,0.f}};

  const float* xrow0 = &Xs[(rtb + 0) * 16 + lane15][0];
  const float* xrow1 = &Xs[(rtb + 1) * 16 + lane15][0];
  const float* hrow  = &Hs[rt * 16 + lane15][0];
  const float* w2be  = W2 + (size_t)e * Hdim * Pdim;

  for (int hk = 0; hk < 512; hk += 64) {
    const int hcol0 = h0 + hk + hg * 16;

    // ---- GEMM1: htile[64,64] = X[64,512] @ W1[e][512, hk..hk+63] ----
    v8f c1[2] = {{0.f,0.f,0.f,0.f,0.f,0.f,0.f,0.f},
                 {0.f,0.f,0.f,0.f,0.f,0.f,0.f,0.f}};
    const float* wp = W1 + ((size_t)e * Sdim + khalf) * Hdim + hcol0 + lane15;
    if (hk + 64 < 512)  // uniform: prefetch next W1 column block
      __builtin_prefetch(W1 + (size_t)e * Sdim * Hdim + h0 + hk + 64, 0, 1);
#pragma unroll 4
    for (int k = 0; k < Sdim; k += 4) {
      v2f bb;
      bb[0] = wp[0]; bb[1] = wp[Hdim];          // B: N=lane15, K rows
      wp += 4 * Hdim;
      v2f a0, a1;
      const float2 f0 = *(const float2*)(&xrow0[k + khalf]);
      const float2 f1 = *(const float2*)(&xrow1[k + khalf]);
      a0[0] = f0.x; a0[1] = f0.y;               // A: M=lane15, K=k+khalf(+1)
      a1[0] = f1.x; a1[1] = f1.y;
      c1[0] = wmma_f32x4(a0, bb, c1[0]);        // B fragment reused x2
      c1[1] = wmma_f32x4(a1, bb, c1[1]);
    }
    // bias + exact GELU, re-layout C->A via LDS
    const float b1v = b1[e * Hdim + hcol0 + lane15];
#pragma unroll
    for (int rtI = 0; rtI < 2; ++rtI) {
      const int mrow = (rtb + rtI) * 16 + (lane >> 4) * 8;
#pragma unroll
      for (int v = 0; v < 8; ++v) {
        const float xv = c1[rtI][v] + b1v;
        const float g = 0.5f * xv * (1.f + erff(xv * 0.70710678118654752f));
        Hs[mrow + v][hg * 16 + lane15] = g;
      }
    }
    __syncthreads();

    // ---- GEMM2: acc += htile[64,64] @ W2[e][hk..hk+63, P] ----
#pragma unroll 4
    for (int kk = 0; kk < 64; kk += 4) {
      v2f a;
      const float2 af = *(const float2*)(&hrow[kk + khalf]);
      a[0] = af.x; a[1] = af.y;
      const float* w2k = w2be + (size_t)(h0 + hk + kk + khalf) * Pdim + lane15;
#pragma unroll
      for (int j = 0; j < 3; ++j) {             // A fragment reused x3
        const int pcol = (ptb + 2 * j) * 16;
        v2f bb;
        bb[0] = w2k[pcol]; bb[1] = w2k[pcol + Pdim];
        acc[j] = wmma_f32x4(a, bb, acc[j]);
      }
    }
    __syncthreads();
  }

  // gate-scale + accumulate into out[b,f,p]
  const int mbase = rt * 16 + (lane >> 4) * 8;
#pragma unroll
  for (int v = 0; v < 8; ++v) {
    const int r = r0 + mbase + v;
    const float g = gates[r * 8 + e];
    const int f = r >> 6, b = r & 63;
    float* orow = out + (size_t)(b * Fdim + f) * Pdim;
#pragma unroll
    for (int j = 0; j < 3; ++j) {
      const int pcol = (ptb + 2 * j) * 16 + lane15;
      atomicAdd(&orow[pcol], g * acc[j][v]);
    }
  }
}

// ---------------------------------------------------------------------------
extern "C" void kernel_launch(void* const* d_in, const int* in_sizes, int n_in,
                              void* d_out, int out_size, void* d_ws,
                              size_t ws_size, hipStream_t stream) {
  (void)in_sizes; (void)n_in; (void)out_size; (void)ws_size;
  const float* x  = (const float*)d_in[0];
  const float* te = (const float*)d_in[1];
  const float* Wg = (const float*)d_in[2];
  const float* bg = (const float*)d_in[3];
  const float* W1 = (const float*)d_in[4];
  const float* b1 = (const float*)d_in[5];
  const float* W2 = (const float*)d_in[6];
  const float* b2 = (const float*)d_in[7];
  float* out   = (float*)d_out;
  float* gates = (float*)d_ws;   // FB*E floats = 64 KB scratch

  gates_kernel<<<FB / 8, 256, 0, stream>>>(te, Wg, bg, gates);
  init_kernel<<<OUTN / 256, 256, 0, stream>>>(gates, b2, out);
  loss_kernel<<<1, 256, 0, stream>>>(gates, out);
  moe_kernel<<<dim3(FB / 64, Edim, Hdim / 512), 256, 0, stream>>>(
      x, W1, b1, W2, b2, gates, out);
}